// CenterLoss_3075196584109
// MI455X (gfx1250) — compile-verified
//
#include <hip/hip_runtime.h>
#include <hip/hip_bf16.h>

typedef float v2f __attribute__((ext_vector_type(2)));
typedef float v8f __attribute__((ext_vector_type(8)));

#define FEAT        128
#define BATCH       1024
#define NUM_CLASSES 100000
#define CLAMP_MIN   1e-12f
#define CLAMP_MAX   1e12f

// One wave (32 threads) handles a tile of 16 samples.
// Computes D = Delta * Delta^T via V_WMMA_F32_16X16X4_F32 accumulation over K,
// where Delta[m][k] = x[sample_m][k] - centers[label_m][k].
// diag(D)[m] = squared distance of sample m. Clamp, reduce, emit one partial.
__global__ __launch_bounds__(32) void center_dist_kernel(
    const float* __restrict__ x,
    const int*   __restrict__ label,
    const float* __restrict__ centers,
    float*       __restrict__ partial)
{
    const int lane   = threadIdx.x;          // 0..31
    const int tile   = blockIdx.x;           // 0..63  (64 tiles * 16 = 1024)
    const int m      = lane & 15;            // sample within tile (A/B row id)
    const int sample = tile * 16 + m;
    const int kbase  = (lane >> 4) << 1;     // lanes 0-15 -> K+{0,1}; 16-31 -> K+{2,3}

    const int lab = label[sample];
    const float* __restrict__ xrow = x + sample * FEAT;
    const float* __restrict__ crow = centers + (long long)lab * FEAT;

    v8f acc = {0.f, 0.f, 0.f, 0.f, 0.f, 0.f, 0.f, 0.f};

    // K loop: 128 features in chunks of 4 -> 32 WMMA f32 16x16x4 accumulations.
    // A (16x4) and B (4x16 = A^T) have identical per-lane register images for
    // this symmetric Gram product, so the same v2f feeds both operands.
#pragma unroll 8
    for (int k0 = 0; k0 < FEAT; k0 += 4) {
        const int k = k0 + kbase;
        v2f xv = *(const v2f*)(xrow + k);   // 8B-aligned: rows are 512B
        v2f cv = *(const v2f*)(crow + k);
        v2f d;
        d.x = xv.x - cv.x;
        d.y = xv.y - cv.y;
        // 8 args: (neg_a, A, neg_b, B, c_mod, C, reuse_a, reuse_b)
        acc = __builtin_amdgcn_wmma_f32_16x16x4_f32(
            false, d, false, d, (short)0, acc, false, false);
    }

    // Diagonal extraction from 16x16 f32 C/D layout:
    //   lane L (<16):  N=L, M=v        -> diag at v=L, valid only L<8
    //   lane L (>=16): N=L-16, M=v+8   -> diag at v=L-24, valid only L>=24
    const int  idx   = (lane < 8) ? lane : (lane - 24);
    const bool valid = (lane < 8) || (lane >= 24);
    float comp = acc[0];
    comp = (idx == 1) ? acc[1] : comp;
    comp = (idx == 2) ? acc[2] : comp;
    comp = (idx == 3) ? acc[3] : comp;
    comp = (idx == 4) ? acc[4] : comp;
    comp = (idx == 5) ? acc[5] : comp;
    comp = (idx == 6) ? acc[6] : comp;
    comp = (idx == 7) ? acc[7] : comp;

    // Per-entry clamp (matches reference: clip applied before the sum).
    float dsq = valid ? fminf(fmaxf(comp, CLAMP_MIN), CLAMP_MAX) : 0.0f;

    // Wave32 butterfly reduction over the 16 valid diagonal values.
#pragma unroll
    for (int off = 16; off > 0; off >>= 1)
        dsq += __shfl_xor(dsq, off, 32);

    if (lane == 0)
        partial[tile] = dsq;
}

// Deterministic final reduction of the 64 tile partials (single wave).
__global__ __launch_bounds__(32) void center_reduce_kernel(
    const float* __restrict__ partial,
    float*       __restrict__ out)
{
    const int lane = threadIdx.x;
    float s = partial[lane] + partial[lane + 32];
#pragma unroll
    for (int off = 16; off > 0; off >>= 1)
        s += __shfl_xor(s, off, 32);
    if (lane == 0) {
        // All B*(C-1) masked-off zeros clamp up to 1e-12 and divide by B:
        // constant term = (C-1) * 1e-12.
        const float mask_clamp_const = (float)((double)(NUM_CLASSES - 1) * 1e-12);
        out[0] = s * (1.0f / (float)BATCH) + mask_clamp_const;
    }
}

extern "C" void kernel_launch(void* const* d_in, const int* in_sizes, int n_in,
                              void* d_out, int out_size, void* d_ws, size_t ws_size,
                              hipStream_t stream)
{
    const float* x       = (const float*)d_in[0];   // [1024, 128] f32
    const int*   label   = (const int*)  d_in[1];   // [1024] int
    const float* centers = (const float*)d_in[2];   // [100000, 128] f32
    float*       out     = (float*)d_out;           // scalar f32
    float*       partial = (float*)d_ws;            // 64 floats of scratch

    center_dist_kernel<<<dim3(BATCH / 16), dim3(32), 0, stream>>>(x, label, centers, partial);
    center_reduce_kernel<<<dim3(1), dim3(32), 0, stream>>>(partial, out);
}